// LinearGCN_36799279793050
// MI455X (gfx1250) — compile-verified
//
#include <hip/hip_runtime.h>

typedef __attribute__((ext_vector_type(2))) float v2f;
typedef __attribute__((ext_vector_type(8))) float v8f;

#define D_FEAT 128
#define N_NODES 100000
#define LDS_STRIDE 132   // 128 + 4 pad: row stride 528B -> 4-bank rotation, no 16-way conflicts

// ---------------------------------------------------------------------------
// Kernel 1: zero the output (harness poisons it with 0xAA)
// ---------------------------------------------------------------------------
__global__ void zero_f4(float4* __restrict__ p, int n4) {
    int i = blockIdx.x * blockDim.x + threadIdx.x;
    if (i < n4) p[i] = make_float4(0.f, 0.f, 0.f, 0.f);
}

// ---------------------------------------------------------------------------
// Kernel 2: Z = H @ W^T  via V_WMMA_F32_16X16X4_F32
//   H: [N_NODES, 128] row-major, W: [128, 128] row-major (W[o][d])
//   Z[n][o] = sum_d H[n][d] * W[o][d]
// One wave computes a 16-row stripe of Z across all 8 column tiles.
// Block = 256 threads = 8 waves = 8 row tiles; W cached in LDS (padded).
// ---------------------------------------------------------------------------
__global__ __launch_bounds__(256) void gemm_hwt_wmma(
    const float* __restrict__ h, const float* __restrict__ w,
    float* __restrict__ z)
{
    __shared__ float wl[D_FEAT * LDS_STRIDE];

    const int t = threadIdx.x;
    // Cooperative load of W into LDS, float4 stores (row stride 132 floats is
    // 16B-aligned: 528 % 16 == 0).
    for (int i4 = t; i4 < (D_FEAT * D_FEAT / 4); i4 += 256) {
        const int row = i4 >> 5;        // 32 float4 per 128-float row
        const int c4  = i4 & 31;
        const float4 val = reinterpret_cast<const float4*>(w)[i4];
        *reinterpret_cast<float4*>(&wl[row * LDS_STRIDE + c4 * 4]) = val;
    }
    __syncthreads();

    const int wave = t >> 5;
    const int lane = t & 31;
    const int m    = lane & 15;         // M (for A) / N (for B) within tile
    const int hi   = lane >> 4;         // selects K pair {0,1} vs {2,3}

    const int rt = blockIdx.x * 8 + wave;           // 16-row tile index
    if (rt * 16 >= N_NODES) return;                 // 6250 tiles total
    const int rowBase = rt * 16;

    v8f acc[8];
    const v8f vzero = {};
    #pragma unroll
    for (int ct = 0; ct < 8; ++ct) acc[ct] = vzero;

    const float* arow = h + (size_t)(rowBase + m) * D_FEAT;

    for (int kb = 0; kb < D_FEAT; kb += 4) {
        const int ko = kb + 2 * hi;                 // even -> 8B aligned
        // A fragment (16x4 f32): VGPR v holds K = 2*hi + v for row m
        const v2f a = *reinterpret_cast<const v2f*>(arow + ko);
        #pragma unroll
        for (int ct = 0; ct < 8; ++ct) {
            // B fragment (4x16 f32): B[k][n] = W[ct*16 + n][k]
            const v2f b = *reinterpret_cast<const v2f*>(
                &wl[(ct * 16 + m) * LDS_STRIDE + ko]);
            acc[ct] = __builtin_amdgcn_wmma_f32_16x16x4_f32(
                /*neg_a=*/false, a, /*neg_b=*/false, b,
                /*c_mod=*/(short)0, acc[ct],
                /*reuse_a=*/false, /*reuse_b=*/false);
        }
    }

    // C/D layout: VGPR r -> M = r + 8*hi, N = m
    #pragma unroll
    for (int ct = 0; ct < 8; ++ct) {
        #pragma unroll
        for (int r = 0; r < 8; ++r) {
            z[(size_t)(rowBase + r + 8 * hi) * D_FEAT + ct * 16 + m] = acc[ct][r];
        }
    }
}

// ---------------------------------------------------------------------------
// Kernel 3: scatter-accumulate: out[row[e]] += val[e] * Z[col[e]]
// One wave per edge; lane l owns features 4l..4l+3 (float4 gather, 4 HW fp
// atomics). Z and out are L2-resident (51.2 MB each vs 192 MB L2).
// ---------------------------------------------------------------------------
__global__ __launch_bounds__(256) void scatter_edges(
    const int* __restrict__ erow, const int* __restrict__ ecol,
    const float* __restrict__ eval, const float* __restrict__ z,
    float* __restrict__ out, int nEdges)
{
    const int wave = threadIdx.x >> 5;
    const int lane = threadIdx.x & 31;
    const int e = blockIdx.x * 8 + wave;
    if (e >= nEdges) return;

    const int   r = erow[e];
    const int   c = ecol[e];
    const float v = eval[e];

    const float4 zc = reinterpret_cast<const float4*>(z + (size_t)c * D_FEAT)[lane];
    float* orow = out + (size_t)r * D_FEAT + lane * 4;

    unsafeAtomicAdd(orow + 0, v * zc.x);   // global_atomic_add_f32
    unsafeAtomicAdd(orow + 1, v * zc.y);
    unsafeAtomicAdd(orow + 2, v * zc.z);
    unsafeAtomicAdd(orow + 3, v * zc.w);
}

// ---------------------------------------------------------------------------
extern "C" void kernel_launch(void* const* d_in, const int* in_sizes, int n_in,
                              void* d_out, int out_size, void* d_ws, size_t ws_size,
                              hipStream_t stream) {
    const float* h    = (const float*)d_in[0];   // [100000,128] f32
    const int*   erow = (const int*)  d_in[1];   // [1.6M] i32
    const int*   ecol = (const int*)  d_in[2];   // [1.6M] i32
    const float* ev   = (const float*)d_in[3];   // [1.6M] f32
    const float* w    = (const float*)d_in[4];   // [128,128] f32
    float* out = (float*)d_out;                  // [100000,128] f32
    float* z   = (float*)d_ws;                   // scratch: [100000,128] f32 = 51.2 MB

    const int nEdges = in_sizes[1];

    // 1) out = 0
    const int n4 = out_size / 4;
    zero_f4<<<(n4 + 255) / 256, 256, 0, stream>>>((float4*)out, n4);

    // 2) Z = H @ W^T   (WMMA f32 16x16x4)
    const int nRowTiles = (N_NODES + 15) / 16;           // 6250
    const int gemmBlocks = (nRowTiles + 7) / 8;          // 8 waves/block
    gemm_hwt_wmma<<<gemmBlocks, 256, 0, stream>>>(h, w, z);

    // 3) out[r] += v * Z[c]
    scatter_edges<<<(nEdges + 7) / 8, 256, 0, stream>>>(erow, ecol, ev, z, out, nEdges);
}